// GNN_41867341201885
// MI455X (gfx1250) — compile-verified
//
#include <hip/hip_runtime.h>

typedef __attribute__((ext_vector_type(2))) float v2f;
typedef __attribute__((ext_vector_type(8))) float v8f;

#define NN 16384      // N_NODES
#define HID 64        // HIDDEN
#define NE 524288     // N_EDGES

// ---------------- degree / normalization ----------------
__global__ void k_deg_init(float* __restrict__ deg) {
    int i = blockIdx.x * blockDim.x + threadIdx.x;
    if (i < NN) deg[i] = 1.0f;                 // self-loop weight
}

__global__ void k_deg_edges(const long long* __restrict__ ei,
                            const float* __restrict__ w,
                            float* __restrict__ deg) {
    int e = blockIdx.x * blockDim.x + threadIdx.x;
    if (e < NE) {
        long long d = ei[NE + e];
        atomicAdd(&deg[d], w[e]);
    }
}

__global__ void k_dinv(float* __restrict__ deg) {
    int i = blockIdx.x * blockDim.x + threadIdx.x;
    if (i < NN) {
        float v = deg[i];
        deg[i] = (v > 0.0f) ? rsqrtf(v) : 0.0f;
    }
}

__global__ void k_normw(const long long* __restrict__ ei,
                        const float* __restrict__ w,
                        const float* __restrict__ dinv,
                        float* __restrict__ nw) {
    int e = blockIdx.x * blockDim.x + threadIdx.x;
    if (e < NE) {
        long long s = ei[e];
        long long d = ei[NE + e];
        nw[e] = dinv[s] * w[e] * dinv[d];
    }
}

// ---------------- h1 = x @ W1  via V_WMMA_F32_16X16X4_F32 ----------------
// Wave computes one 16x16 tile. Block = 4 waves = 16 rows x 64 cols. 1024 blocks.
__global__ __launch_bounds__(128) void k_gemm1(const float* __restrict__ x,
                                               const float* __restrict__ W1,
                                               float* __restrict__ h1) {
    const int lane = threadIdx.x & 31;
    const int wave = threadIdx.x >> 5;
    const int rows = blockIdx.x << 4;
    const int cols = wave << 4;
    const int lr   = lane & 15;
    const int hi   = lane >> 4;        // lanes 16-31 hold K+2,K+3
    const int kOff = hi << 1;

    const float* xrow = x  + (size_t)(rows + lr) * NN + kOff;     // A: row lr, K=k+kOff..+1
    const float* wcol = W1 + (size_t)kOff * HID + cols + lr;      // B: col (cols+lr), K=k+kOff..+1

    v8f acc = {};
#pragma unroll 4
    for (int k = 0; k < NN; k += 4) {
        float2 a2 = *(const float2*)(xrow + k);
        v2f A = { a2.x, a2.y };
        v2f B = { wcol[(size_t)k * HID], wcol[(size_t)(k + 1) * HID] };
        acc = __builtin_amdgcn_wmma_f32_16x16x4_f32(
            /*neg_a=*/false, A, /*neg_b=*/false, B,
            /*c_mod=*/(short)0, acc, /*reuse_a=*/false, /*reuse_b=*/false);
    }

    // C/D layout: VGPR v, lanes 0-15 -> M=v, lanes 16-31 -> M=v+8; N = lane&15
    float* out = h1 + (size_t)(rows + (hi ? 8 : 0)) * HID + cols + lr;
#pragma unroll
    for (int v = 0; v < 8; ++v) out[(size_t)v * HID] = acc[v];
}

// ---------------- aggregation layer 1 ----------------
__global__ void k_agg_init(const float* __restrict__ h1,
                           const float* __restrict__ dinv,
                           float* __restrict__ agg) {
    int i = blockIdx.x * blockDim.x + threadIdx.x;   // over NN*HID
    float dv = dinv[i >> 6];
    agg[i] = dv * dv * h1[i];                        // self-loop term
}

__global__ void k_agg_edges(const long long* __restrict__ ei,
                            const float* __restrict__ nw,
                            const float* __restrict__ h1,
                            float* __restrict__ agg) {
    unsigned tid = blockIdx.x * blockDim.x + threadIdx.x;   // over NE*HID
    unsigned e = tid >> 6;
    unsigned c = tid & 63;
    long long s = ei[e];
    long long d = ei[NE + e];
    atomicAdd(&agg[(size_t)d * HID + c], nw[e] * h1[(size_t)s * HID + c]);
}

__global__ void k_relu_bias(const float* __restrict__ agg,
                            const float* __restrict__ b1,
                            float* __restrict__ h) {
    int i = blockIdx.x * blockDim.x + threadIdx.x;   // over NN*HID
    float v = agg[i] + b1[i & 63];
    h[i] = v > 0.0f ? v : 0.0f;
}

// ---------------- h2 = h @ W2 (64 -> 1), one wave per row ----------------
__global__ void k_gemm2(const float* __restrict__ h,
                        const float* __restrict__ W2,
                        float* __restrict__ h2) {
    int row  = blockIdx.x * 8 + (threadIdx.x >> 5);
    int lane = threadIdx.x & 31;
    const float* hr = h + (size_t)row * HID;
    float s = hr[lane] * W2[lane] + hr[32 + lane] * W2[32 + lane];
#pragma unroll
    for (int off = 16; off > 0; off >>= 1) s += __shfl_down(s, off, 32);
    if (lane == 0) h2[row] = s;
}

// ---------------- aggregation layer 2 + bias ----------------
__global__ void k_out_init(const float* __restrict__ h2,
                           const float* __restrict__ dinv,
                           const float* __restrict__ b2,
                           float* __restrict__ out) {
    int i = blockIdx.x * blockDim.x + threadIdx.x;
    if (i < NN) {
        float dv = dinv[i];
        out[i] = b2[0] + dv * dv * h2[i];
    }
}

__global__ void k_out_edges(const long long* __restrict__ ei,
                            const float* __restrict__ nw,
                            const float* __restrict__ h2,
                            float* __restrict__ out) {
    int e = blockIdx.x * blockDim.x + threadIdx.x;
    if (e < NE) {
        long long s = ei[e];
        long long d = ei[NE + e];
        atomicAdd(&out[d], nw[e] * h2[s]);
    }
}

extern "C" void kernel_launch(void* const* d_in, const int* in_sizes, int n_in,
                              void* d_out, int out_size, void* d_ws, size_t ws_size,
                              hipStream_t stream) {
    const float*     x  = (const float*)d_in[0];       // [NN, NN]
    const long long* ei = (const long long*)d_in[1];   // [2, NE] int64
    const float*     w  = (const float*)d_in[2];       // [NE]
    const float*     W1 = (const float*)d_in[3];       // [NN, HID]
    const float*     b1 = (const float*)d_in[4];       // [HID]
    const float*     W2 = (const float*)d_in[5];       // [HID, 1]
    const float*     b2 = (const float*)d_in[6];       // [1]
    float*           out = (float*)d_out;              // [NN]

    // workspace layout (floats): dinv | norm_w | h1 | agg | h2
    float* dinv = (float*)d_ws;                // NN
    float* nw   = dinv + NN;                   // NE
    float* h1   = nw + NE;                     // NN*HID
    float* agg  = h1 + (size_t)NN * HID;       // NN*HID
    float* h2   = agg + (size_t)NN * HID;      // NN

    // normalization
    k_deg_init <<<NN / 256, 256, 0, stream>>>(dinv);
    k_deg_edges<<<NE / 256, 256, 0, stream>>>(ei, w, dinv);
    k_dinv     <<<NN / 256, 256, 0, stream>>>(dinv);
    k_normw    <<<NE / 256, 256, 0, stream>>>(ei, w, dinv, nw);

    // layer 1
    k_gemm1    <<<NN / 16, 128, 0, stream>>>(x, W1, h1);
    k_agg_init <<<(NN * HID) / 256, 256, 0, stream>>>(h1, dinv, agg);
    k_agg_edges<<<(NE / 256) * HID, 256, 0, stream>>>(ei, nw, h1, agg);
    k_relu_bias<<<(NN * HID) / 256, 256, 0, stream>>>(agg, b1, h1);  // h reuses h1 buffer

    // layer 2
    k_gemm2    <<<NN / 8, 256, 0, stream>>>(h1, W2, h2);
    k_out_init <<<NN / 256, 256, 0, stream>>>(h2, dinv, b2, out);
    k_out_edges<<<NE / 256, 256, 0, stream>>>(ei, nw, h2, out);
}